// Repro_87402584474057
// MI455X (gfx1250) — compile-verified
//
#include <hip/hip_runtime.h>
#include <math.h>

typedef float v2f __attribute__((ext_vector_type(2)));
typedef float v8f __attribute__((ext_vector_type(8)));

#define TWO_PI 6.28318530717958647692f

// ---------------------------------------------------------------------------
// Kernel 1: base = primals_1 + 0.975 * primals_3   (built once, reused 3x)
// ---------------------------------------------------------------------------
__global__ void build_base_kernel(const float* __restrict__ p1,
                                  const float* __restrict__ p3,
                                  float* __restrict__ mem, int n4) {
  int i = blockIdx.x * blockDim.x + threadIdx.x;
  if (i < n4) {
    const float4* a = (const float4*)p1;
    const float4* b = (const float4*)p3;
    float4 va = a[i], vb = b[i];
    float4 r;
    r.x = fmaf(vb.x, 0.975f, va.x);
    r.y = fmaf(vb.y, 0.975f, va.y);
    r.z = fmaf(vb.z, 0.975f, va.z);
    r.w = fmaf(vb.w, 0.975f, va.w);
    ((float4*)mem)[i] = r;
  }
}

// ---------------------------------------------------------------------------
// Kernel 2: per-stage scatter. Positions identical across stages, so a plain
// overwrite of the same 2048 slots refreshes mem for the new stage.
// mem[b,d,e] = table[b, p6] + 0.975 * p3[b,d,e]
// ---------------------------------------------------------------------------
__global__ void scatter_kernel(float* __restrict__ mem,
                               const float* __restrict__ table,
                               const float* __restrict__ p3,
                               const int* __restrict__ p5,
                               const int* __restrict__ p6,
                               const int* __restrict__ p7,
                               const int* __restrict__ p8, int K) {
  int k = blockIdx.x * blockDim.x + threadIdx.x;
  if (k < K) {
    int b = p5[k];
    int j = p6[k];
    int d = p7[k];
    int e = p8[k];
    float val = table[b * 256 + j];
    int idx = b * 65536 + d * 256 + e;
    mem[idx] = fmaf(p3[idx], 0.975f, val);
  }
}

// ---------------------------------------------------------------------------
// Kernel 3: WMMA GEMM stage.
//   att[h,b,e] = sum_d q[h,b,d] * mem[b,d,e]   (M=12 padded to 16, K=256, N=256)
//   out = att / m  (mode 0)  or  att * m  (mode 1),  m[e] = sin(2*pi*e*f+ph)^2*0.1+0.95
// One wave (32 lanes) per 16x16 output tile. grid = (16 N-tiles, 6 batches).
//
// V_WMMA_F32_16X16X4_F32 fragment layout (wave32):
//   A (16x4):  lanes 0-15 hold M=lane, K={k,k+1} in v0/v1; lanes 16-31 hold K={k+2,k+3}
//   B (4x16):  lanes 0-15 hold N=lane, K={k,k+1} in v0/v1; lanes 16-31 hold K={k+2,k+3}
//   C/D:       VGPR r: M=r (lanes 0-15) / M=r+8 (lanes 16-31), N=lane&15
// ---------------------------------------------------------------------------
__global__ void gemm_stage_kernel(const float* __restrict__ q,
                                  const float* __restrict__ mem,
                                  const float* __restrict__ freq,
                                  const float* __restrict__ phase,
                                  float* __restrict__ out, int mode) {
  const int lane = threadIdx.x;        // 0..31
  const int nt   = blockIdx.x;         // 0..15  (N tile)
  const int b    = blockIdx.y;         // 0..5   (batch)

  const int half = lane >> 4;          // 0 or 1
  const int m    = lane & 15;          // row within tile (A) / col (B)
  const int col  = (nt << 4) + m;      // output column e
  const int koff = half << 1;          // 0 or 2

  const bool  mvalid = (m < 12);
  const float aflag  = mvalid ? 1.0f : 0.0f;
  // padded rows read row 0 and get zeroed by aflag
  const float* __restrict__ Arow  = q + (((mvalid ? m : 0) * 6 + b) << 8);
  const float* __restrict__ Bbase = mem + b * 65536 + col;

  v8f acc = {0.f, 0.f, 0.f, 0.f, 0.f, 0.f, 0.f, 0.f};

  // Outer loop: 16 K-rows per step, one near-scope prefetch 16 rows ahead.
  // Overshoot past the batch slice lands in still-allocated workspace
  // (x1/x2), and speculative prefetches are dropped on bad translation,
  // so no guard is needed -> branchless inner loop.
  for (int k = 0; k < 256; k += 16) {
    __builtin_prefetch(Bbase + (size_t)(k + 16 + koff) * 256, 0, 3);

#pragma unroll
    for (int kk = 0; kk < 16; kk += 4) {
      const int d0 = k + kk + koff;

      // A fragment: two consecutive K values for this lane's row (8B aligned)
      v2f a = *(const v2f*)(Arow + d0);
      a.x *= aflag;
      a.y *= aflag;

      // B fragment: two K-rows, lane-contiguous columns (coalesced b32 loads)
      v2f bb;
      bb.x = Bbase[(size_t)d0 * 256];
      bb.y = Bbase[(size_t)(d0 + 1) * 256];

      acc = __builtin_amdgcn_wmma_f32_16x16x4_f32(
          /*neg_a=*/false, a, /*neg_b=*/false, bb,
          /*c_mod=*/(short)0, acc, /*reuse_a=*/false, /*reuse_b=*/false);
    }
  }

  // modulation for this column
  const float s   = __sinf(fmaf((float)col * TWO_PI, freq[0], phase[0]));
  const float mod = fmaf(s * s, 0.1f, 0.95f);
  const float f   = (mode == 0) ? (1.0f / mod) : mod;

#pragma unroll
  for (int r = 0; r < 8; ++r) {
    const int M = r + (half << 3);
    if (M < 12) {
      out[((M * 6 + b) << 8) + col] = acc[r] * f;
    }
  }
}

// ---------------------------------------------------------------------------
// Host-side orchestration (stream-ordered, graph-capture safe).
// ws layout: [ mem: 6*256*256 f32 | x1: 12*6*256 f32 | x2: 12*6*256 f32 ]
// ---------------------------------------------------------------------------
extern "C" void kernel_launch(void* const* d_in, const int* in_sizes, int n_in,
                              void* d_out, int out_size, void* d_ws, size_t ws_size,
                              hipStream_t stream) {
  const float* p1  = (const float*)d_in[0];   // [6,256,256]
  const float* p2  = (const float*)d_in[1];   // [12,6,256]
  const float* p3  = (const float*)d_in[2];   // [6,256,256]
  const float* p4  = (const float*)d_in[3];   // [6,256]
  const int*   p5  = (const int*)d_in[4];     // [K]
  const int*   p6  = (const int*)d_in[5];
  const int*   p7  = (const int*)d_in[6];
  const int*   p8  = (const int*)d_in[7];
  const float* p9  = (const float*)d_in[8];   // freq1
  const float* p10 = (const float*)d_in[9];   // phase1
  const float* p11 = (const float*)d_in[10];  // table2
  const float* p12 = (const float*)d_in[11];  // freq2
  const float* p13 = (const float*)d_in[12];  // phase2
  const float* p14 = (const float*)d_in[13];  // table3
  const float* p15 = (const float*)d_in[14];  // freq3
  const float* p16 = (const float*)d_in[15];  // phase3

  const int K = in_sizes[4];                  // 2048

  float* mem = (float*)d_ws;                  // 6*256*256
  float* x1  = mem + 6 * 256 * 256;           // 12*6*256
  float* x2  = x1 + 12 * 6 * 256;             // 12*6*256
  float* yout = (float*)d_out;                // 12*6*256

  // base = p1 + 0.975*p3 (once; later stages only rewrite the 2048 slots)
  {
    const int n4 = (6 * 256 * 256) / 4;
    build_base_kernel<<<(n4 + 255) / 256, 256, 0, stream>>>(p1, p3, mem, n4);
  }

  const dim3 gemm_grid(16, 6);
  const dim3 gemm_block(32);
  const int scat_blocks = (K + 255) / 256;

  // stage 1: q = p2, out = x1 = att / m1
  scatter_kernel<<<scat_blocks, 256, 0, stream>>>(mem, p4, p3, p5, p6, p7, p8, K);
  gemm_stage_kernel<<<gemm_grid, gemm_block, 0, stream>>>(p2, mem, p9, p10, x1, 0);

  // stage 2: q = x1, out = x2 = att * m2
  scatter_kernel<<<scat_blocks, 256, 0, stream>>>(mem, p11, p3, p5, p6, p7, p8, K);
  gemm_stage_kernel<<<gemm_grid, gemm_block, 0, stream>>>(x1, mem, p12, p13, x2, 1);

  // stage 3: q = x2, out = d_out = att / m3
  scatter_kernel<<<scat_blocks, 256, 0, stream>>>(mem, p14, p3, p5, p6, p7, p8, K);
  gemm_stage_kernel<<<gemm_grid, gemm_block, 0, stream>>>(x2, mem, p15, p16, yout, 0);
}